// Q2n_60430189855128
// MI455X (gfx1250) — compile-verified
//
#include <hip/hip_runtime.h>
#include <math.h>

typedef float v2f __attribute__((ext_vector_type(2)));
typedef float v8f __attribute__((ext_vector_type(8)));
typedef int v4i __attribute__((ext_vector_type(4)));

#define HW 2048
#define PLANE ((size_t)HW * (size_t)HW)
#define NBX 64
#define CSTRIDE 260        // dwords per channel plane in LDS (256 px + 4 pad)
#define CHUNK_PIX 256      // 8 rows of 32
#define NCHUNK 4

// Cayley-Dickson (C=8) bilinear table, derived from the reference recursion:
// contribution of Cov[i][j] goes to channel TCt[i][j] with sign TSt[i][j].
__device__ __constant__ const int TCt[8][8] = {
    {0, 1, 2, 3, 4, 5, 6, 7}, {1, 0, 3, 2, 5, 4, 7, 6},
    {2, 3, 0, 1, 6, 7, 4, 5}, {3, 2, 1, 0, 7, 6, 5, 4},
    {4, 5, 6, 7, 0, 1, 2, 3}, {5, 4, 7, 6, 1, 0, 3, 2},
    {6, 7, 4, 5, 2, 3, 0, 1}, {7, 6, 5, 4, 3, 2, 1, 0}};
__device__ __constant__ const float TSt[8][8] = {
    {+1, +1, +1, -1, +1, -1, -1, +1}, {+1, -1, -1, -1, -1, -1, -1, -1},
    {+1, +1, -1, +1, -1, +1, -1, +1}, {-1, +1, -1, -1, +1, +1, -1, -1},
    {+1, +1, +1, -1, -1, +1, +1, -1}, {-1, +1, -1, -1, -1, -1, +1, +1},
    {-1, +1, +1, +1, -1, -1, -1, -1}, {+1, +1, -1, +1, +1, -1, +1, -1}};

#define AS1P(p) ((__attribute__((address_space(1))) v4i*)(p))
#define AS3P(p) ((__attribute__((address_space(3))) v4i*)(p))

__global__ __launch_bounds__(32) void q2n_block_kernel(
    const float* __restrict__ outputs, const float* __restrict__ labels,
    float* __restrict__ qmap) {
  // Double-buffered channel-plane tiles: buf[sel][c * CSTRIDE + pixel]
  __shared__ __align__(16) float buf[2][16 * CSTRIDE];  // 2 x 16640 B
  __shared__ float Ssum[16];
  __shared__ float G[16][16];

  const int lane = threadIdx.x;  // wave32: one wave per 32x32 block
  const int b = blockIdx.x;
  const int bx = b & (NBX - 1);
  const int by = b >> 6;

  // ---- async staging: 32 x GLOBAL_LOAD_ASYNC_TO_LDS_B128 per 256-px chunk.
  // Lane l fetches 4 contiguous pixels (16 B) of one channel row and the async
  // unit drops them at the matching 16 B-aligned LDS spot (layout needs no
  // in-register transpose).
  auto stage = [&](int ch, int bsel) {
    const int rsub = lane >> 3;          // row within 4-row group
    const int col = (lane & 7) << 2;     // 4-pixel column start
    const size_t gb =
        (size_t)(by * 32 + ch * 8) * HW + (size_t)(bx * 32 + col);
#pragma unroll
    for (int k = 0; k < 2; ++k) {        // two 128-px groups per chunk
      const size_t go = gb + (size_t)(k * 4 + rsub) * HW;
      const int lo = k * 128 + lane * 4;  // pixel index in chunk
      float* lb = &buf[bsel][lo];
#pragma unroll
      for (int c = 0; c < 8; ++c) {
        __builtin_amdgcn_global_load_async_to_lds_b128(
            AS1P(labels + (size_t)c * PLANE + go), AS3P(lb + c * CSTRIDE), 0,
            0);
        __builtin_amdgcn_global_load_async_to_lds_b128(
            AS1P(outputs + (size_t)c * PLANE + go),
            AS3P(lb + (8 + c) * CSTRIDE), 0, 0);
      }
    }
  };

  v8f acc = {0.f, 0.f, 0.f, 0.f, 0.f, 0.f, 0.f, 0.f};
  float s = 0.f;  // per-lane raw sum of this lane's (extended) channel
  const int cc = lane & 15;
  const int half = lane >> 4;

  stage(0, 0);
  for (int ch = 0; ch < NCHUNK; ++ch) {
    if (ch + 1 < NCHUNK) {
      stage(ch + 1, (ch + 1) & 1);                 // prefetch next chunk
      __builtin_amdgcn_s_wait_asynccnt(32);        // current chunk landed
    } else {
      __builtin_amdgcn_s_wait_asynccnt(0);
    }
    asm volatile("" ::: "memory");
    // ---- Gram accumulation: G += X^T X via V_WMMA_F32_16X16X4_F32.
    // A (16x4) and B (4x16) register images coincide for this symmetric
    // product; one conflict-free ds_load_b64 feeds each WMMA.
    const float* tptr = &buf[ch & 1][cc * CSTRIDE + 2 * half];
#pragma unroll 4
    for (int it = 0; it < CHUNK_PIX / 4; ++it) {
      v2f ab = *(const v2f*)tptr;  // pixels k0+2*half, k0+2*half+1
      v2f r;
      r.x = rintf(ab.x);  // jnp.round = round-half-even
      r.y = rintf(ab.y);
      s += r.x + r.y;
      acc = __builtin_amdgcn_wmma_f32_16x16x4_f32(false, r, false, r, (short)0,
                                                  acc, false, false);
      tptr += 4;
    }
  }

  // halves cover disjoint pixel sets of the same channel
  float stot = s + __shfl_xor(s, 16, 32);
  if (lane < 16) Ssum[lane] = stot;
  // C/D layout: VGPR v -> row v + 8*half, col = lane&15
#pragma unroll
  for (int v = 0; v < 8; ++v) G[half * 8 + v][cc] = acc[v];
  __syncthreads();

  if (lane == 0) {
    const float n = 1024.f;
    const float temp = n / (n - 1.f);
    float r1[8], d1[8], r2[8], d2[8], m1[8], m2[8];
    float meanq1 = 0.f, meanq2 = 0.f, mq1m2 = 0.f, mq2m2 = 0.f;
#pragma unroll
    for (int c = 0; c < 8; ++c) {
      float Sx = Ssum[c], Sy = Ssum[8 + c];
      float Qxx = G[c][c], Qyy = G[8 + c][8 + c];
      float m = Sx / n;
      float ssd = fmaxf(Qxx - n * m * m, 0.f);  // sum((x-m)^2), exact for ints
      float t = sqrtf(ssd / (n - 1.f));
      float ts = (t == 0.f) ? 1e-10f : t;
      float rr1 = 1.f / ts;
      float dd1 = 1.f - m * rr1;
      float sgn = (c == 0) ? 1.f : -1.f;
      bool mz = (m == 0.f);
      float rr2 = sgn * (mz ? 1.f : rr1);
      float dd2 = sgn * (mz ? (1.f - m) : (1.f - m * rr1));
      r1[c] = rr1; d1[c] = dd1; r2[c] = rr2; d2[c] = dd2;
      float mm1 = m * rr1 + dd1;  // mean(im1n)
      float my = Sy / n;
      float mm2 = my * rr2 + dd2;  // mean(im2n)
      m1[c] = mm1; m2[c] = mm2;
      mq1m2 += mm1 * mm1;
      mq2m2 += mm2 * mm2;
      // sum(im{1,2}n^2) via affine transform of raw moments
      meanq1 += rr1 * rr1 * Qxx + 2.f * rr1 * dd1 * Sx + n * dd1 * dd1;
      meanq2 += rr2 * rr2 * Qyy + 2.f * rr2 * dd2 * Sy + n * dd2 * dd2;
    }
    meanq1 /= n;
    meanq2 /= n;
    float term2 = sqrtf(mq1m2) * sqrtf(mq2m2);
    float term4 = mq1m2 + mq2m2;
    float term3 = temp * meanq1 + temp * meanq2 - temp * term4;
    float mean_bias = 2.f * term2 / term4;
    float q2n;
    if (term3 == 0.f) {
      q2n = fabsf(mean_bias);  // q_zero: only last channel = mean_bias
    } else {
      float cbm = 2.f / term3;
      float qc[8] = {0.f, 0.f, 0.f, 0.f, 0.f, 0.f, 0.f, 0.f};
#pragma unroll
      for (int i = 0; i < 8; ++i) {
#pragma unroll
        for (int j = 0; j < 8; ++j) {
          float P = G[i][8 + j];  // sum_p raw_x_i * raw_y_j
          float Pn = r1[i] * r2[j] * P + r1[i] * d2[j] * Ssum[i] +
                     d1[i] * r2[j] * Ssum[8 + j] + n * d1[i] * d2[j];
          float cov = Pn / n - m1[i] * m2[j];  // mean(qu) - qm contribution
          qc[TCt[i][j]] += TSt[i][j] * cov;
        }
      }
      float scale = temp * mean_bias * cbm;
      float s2 = 0.f;
#pragma unroll
      for (int c = 0; c < 8; ++c) {
        float q = qc[c] * scale;
        s2 += q * q;
      }
      q2n = sqrtf(s2);
    }
    qmap[b] = q2n;
  }
}

__global__ __launch_bounds__(256) void q2n_mean_kernel(
    float* __restrict__ d_out) {
  __shared__ float red[256];
  int t = threadIdx.x;
  float s = 0.f;
#pragma unroll 4
  for (int i = t; i < 4096; i += 256) s += d_out[1 + i];
  red[t] = s;
  __syncthreads();
#pragma unroll
  for (int off = 128; off > 0; off >>= 1) {
    if (t < off) red[t] += red[t + off];
    __syncthreads();
  }
  if (t == 0) d_out[0] = red[0] * (1.f / 4096.f);
}

extern "C" void kernel_launch(void* const* d_in, const int* in_sizes, int n_in,
                              void* d_out, int out_size, void* d_ws,
                              size_t ws_size, hipStream_t stream) {
  (void)in_sizes; (void)n_in; (void)d_ws; (void)ws_size; (void)out_size;
  const float* outputs = (const float*)d_in[0];
  const float* labels = (const float*)d_in[1];
  float* out = (float*)d_out;
  q2n_block_kernel<<<dim3(64 * 64), dim3(32), 0, stream>>>(outputs, labels,
                                                           out + 1);
  q2n_mean_kernel<<<dim3(1), dim3(256), 0, stream>>>(out);
}